// WindowAttention_52639119180081
// MI455X (gfx1250) — compile-verified
//
#include <hip/hip_runtime.h>
#include <hip/hip_bf16.h>
#include <math.h>

// ---------------------------------------------------------------------------
// Swin-v2 window attention, fused for MI455X (gfx1250, wave32, WMMA).
// fp32 end-to-end using V_WMMA_F32_16X16X4_F32 (workload is HBM-bound at
// 23.3 TB/s; fp32 matrix compute is ~100x faster than the memory stream).
// (a) bank-conflict-free LDS layout for q/k fragments (row stride 36),
// (b) async global->LDS staging via GLOBAL_LOAD_ASYNC_TO_LDS_B128 inline asm
//     (the clang builtin has a different, undocumented prototype: its first
//     parameter is 'v4i*', so we use the ISA mnemonic directly) +
//     s_wait_asynccnt before the barrier.
// ---------------------------------------------------------------------------

typedef __attribute__((ext_vector_type(2))) float v2f;
typedef __attribute__((ext_vector_type(8))) float v8f;

#define WIN 12
#define NTOK 144          // WIN*WIN
#define HEADS 16
#define HDIM 32
#define DIM 512           // HEADS*HDIM
#define BWIN 512          // number of windows*batch
#define NW 64             // windows per image (mask dim)
#define TBL 23            // 2*WIN-1
#define SP 148            // padded LDS stride for the 144-wide score matrix
#define SQ 36             // padded LDS row stride for q/k (floats): 36*m16 mod 64
                          // is a full permutation -> conflict-free WMMA frags

// one 16B async copy: global -> LDS (memory -> LDS DMA path, ASYNCcnt-tracked)
__device__ __forceinline__ void async_cp16(const float4* g, float4* l) {
    // VDST = 32-bit LDS byte address, VADDR = 64-bit global address, no SADDR
    unsigned lds_addr = (unsigned)(uintptr_t)l;
    unsigned long long gaddr = (unsigned long long)(uintptr_t)g;
    asm volatile("global_load_async_to_lds_b128 %0, %1, off"
                 :: "v"(lds_addr), "v"(gaddr)
                 : "memory");
}

__device__ __forceinline__ void async_wait0() {
#if __has_builtin(__builtin_amdgcn_s_wait_asynccnt)
    __builtin_amdgcn_s_wait_asynccnt(0);
#else
    asm volatile("s_wait_asynccnt 0x0" ::: "memory");
#endif
}

// ---------------------------------------------------------------------------
// Kernel 1: CPB MLP -> table[(2W-1)^2][H]
// ---------------------------------------------------------------------------
__device__ __forceinline__ float cpb_coord(int i) {
    float r = (float)(i - (WIN - 1));
    float t = r / (float)(WIN - 1) * 8.0f;
    float v = log2f(fabsf(t) + 1.0f) * (1.0f / 3.0f);   // /log2(8)
    return (t > 0.f) ? v : ((t < 0.f) ? -v : 0.f);
}

__global__ void cpb_table_kernel(const float* __restrict__ w1,
                                 const float* __restrict__ b1,
                                 const float* __restrict__ w2,
                                 float* __restrict__ table) {
    __shared__ float acc[HEADS];
    const int i = blockIdx.x / TBL;
    const int j = blockIdx.x % TBL;
    const float c0 = cpb_coord(i);
    const float c1 = cpb_coord(j);
    if (threadIdx.x < HEADS) acc[threadIdx.x] = 0.f;
    __syncthreads();
    float p[HEADS];
#pragma unroll
    for (int o = 0; o < HEADS; ++o) p[o] = 0.f;
    for (int jh = threadIdx.x; jh < 512; jh += blockDim.x) {
        float hv = fmaxf(c0 * w1[2 * jh] + c1 * w1[2 * jh + 1] + b1[jh], 0.f);
#pragma unroll
        for (int o = 0; o < HEADS; ++o) p[o] += hv * w2[o * 512 + jh];
    }
#pragma unroll
    for (int o = 0; o < HEADS; ++o) atomicAdd(&acc[o], p[o]);
    __syncthreads();
    if (threadIdx.x < HEADS)
        table[blockIdx.x * HEADS + threadIdx.x] = acc[threadIdx.x];
}

// ---------------------------------------------------------------------------
// Kernel 2: bias16[h][n][m] = 16*sigmoid(table[rel_index(n,m)][h])
// ---------------------------------------------------------------------------
__global__ void cpb_bias_kernel(const float* __restrict__ table,
                                float* __restrict__ bias16) {
    int e = blockIdx.x * blockDim.x + threadIdx.x;
    const int TOT = HEADS * NTOK * NTOK;
    if (e >= TOT) return;
    int h = e / (NTOK * NTOK);
    int rem = e % (NTOK * NTOK);
    int n = rem / NTOK, m = rem % NTOK;
    int yi = n / WIN, xi = n % WIN;
    int yj = m / WIN, xj = m % WIN;
    int idx = (yi - yj + WIN - 1) * TBL + (xi - xj + WIN - 1);
    float t = table[idx * HEADS + h];
    bias16[e] = 16.0f / (1.0f + __expf(-t));
}

// ---------------------------------------------------------------------------
// Kernel 3: QKV GEMM.  qkv[row][col] = sum_k x[row][k] * w_qkv[col][k] + bias
// One wave computes one 16x16 output tile via 128 x WMMA_F32_16X16X4_F32.
// ---------------------------------------------------------------------------
__global__ void qkv_gemm_kernel(const float* __restrict__ x,
                                const float* __restrict__ w_qkv,
                                const float* __restrict__ q_bias,
                                const float* __restrict__ v_bias,
                                float* __restrict__ qb,
                                float* __restrict__ kb,
                                float* __restrict__ vb) {
    const int wave = threadIdx.x >> 5;
    const int lane = threadIdx.x & 31;
    const int m16 = lane & 15;
    const int koff = (lane >> 4) << 1;   // 0 or 2 (A/B K split across half-waves)
    const int half8 = (lane >> 4) << 3;  // +8 row offset for D upper half

    const int Mbase = blockIdx.y * 16;              // rows in (B*N) = 73728
    const int Nbase = (blockIdx.x * 8 + wave) * 16; // cols in [0,1536)
    const int col = Nbase + m16;

    const float* arow = x + (size_t)(Mbase + m16) * DIM;
    const float* brow = w_qkv + (size_t)col * DIM;

    v8f c = {};
    for (int kk = 0; kk < DIM; kk += 4) {
        v2f a, b;
        a.x = arow[kk + koff];
        a.y = arow[kk + koff + 1];
        b.x = brow[kk + koff];
        b.y = brow[kk + koff + 1];
        c = __builtin_amdgcn_wmma_f32_16x16x4_f32(false, a, false, b,
                                                  (short)0, c, false, false);
    }

    float bias = 0.f;
    const int which = col >> 9;  // 0=q, 1=k, 2=v
    const int cd = col & (DIM - 1);
    if (which == 0) bias = q_bias[cd];
    else if (which == 2) bias = v_bias[cd];
    const int hh = cd >> 5, dd = cd & 31;
    float* dst = (which == 0) ? qb : ((which == 1) ? kb : vb);

#pragma unroll
    for (int r = 0; r < 8; ++r) {
        int row = Mbase + half8 + r;     // global token row
        int b_ = row / NTOK, n = row % NTOK;
        dst[(((size_t)b_ * HEADS + hh) * NTOK + n) * HDIM + dd] = c[r] + bias;
    }
}

// ---------------------------------------------------------------------------
// Kernel 4: fused per-(b,h) attention.
//   - async-stage q,k,v into LDS (q/k padded to stride SQ for conflict-free
//     WMMA fragment reads; v stays dense since its reads are lane-consecutive)
//   - wave32 shuffle-reduce L2 norms; fold exp(min(logit_scale,ln100)) into q
//   - S = q_hat @ k_hat^T via WMMA, +CPB bias +mask, row softmax, O = P @ V
// LDS: 2*144*36 + 144*32 + 144*148 floats = 145152 B (dynamic; <320KB/WGP)
// ---------------------------------------------------------------------------
__global__ void attention_kernel(const float* __restrict__ qbuf,
                                 const float* __restrict__ kbuf,
                                 const float* __restrict__ vbuf,
                                 const float* __restrict__ bias16,
                                 const float* __restrict__ mask,
                                 const float* __restrict__ logit_scale,
                                 float* __restrict__ attn_out) {
    extern __shared__ float smem[];
    float* qs = smem;                  // 144 rows * SQ
    float* ks = qs + NTOK * SQ;        // 144 rows * SQ
    float* vs = ks + NTOK * SQ;        // 144 * 32 (dense)
    float* S = vs + NTOK * HDIM;       // 144 * SP

    const int bh = blockIdx.x;
    const int b_ = bh >> 4;
    const int h = bh & 15;
    const int tid = threadIdx.x;
    const int wave = tid >> 5;
    const int lane = tid & 31;
    const int m16 = lane & 15;
    const int koff = (lane >> 4) << 1;
    const int half8 = (lane >> 4) << 3;

    // ---- async stage q, k, v (float4 granularity; 8 float4 per token row,
    //      q/k rows padded to 9 float4 slots) ----
    {
        const float4* gq = (const float4*)(qbuf + (size_t)bh * NTOK * HDIM);
        const float4* gk = (const float4*)(kbuf + (size_t)bh * NTOK * HDIM);
        const float4* gv = (const float4*)(vbuf + (size_t)bh * NTOK * HDIM);
        float4* lq = (float4*)qs;
        float4* lk = (float4*)ks;
        float4* lv = (float4*)vs;
        for (int i = tid; i < NTOK * HDIM / 4; i += 256) {
            int row = i >> 3, slot = i & 7;
            int pidx = row * (SQ / 4) + slot;   // padded q/k slot
            async_cp16(gq + i, lq + pidx);
            async_cp16(gk + i, lk + pidx);
            async_cp16(gv + i, lv + i);
        }
    }
    async_wait0();
    __syncthreads();

    // ---- l2-normalize rows; fold clamped logit scale into q ----
    const float scale = __expf(fminf(logit_scale[h], 4.605170185988092f)); // ln(100)
    for (int r = wave; r < NTOK; r += 8) {
        float qv = qs[r * SQ + lane];
        float kv = ks[r * SQ + lane];
        float sq = qv * qv, sk = kv * kv;
#pragma unroll
        for (int off = 16; off; off >>= 1) {
            sq += __shfl_xor(sq, off, 32);
            sk += __shfl_xor(sk, off, 32);
        }
        qs[r * SQ + lane] = qv * (scale / fmaxf(sqrtf(sq), 1e-12f));
        ks[r * SQ + lane] = kv * (1.0f / fmaxf(sqrtf(sk), 1e-12f));
    }
    __syncthreads();

    // ---- S = q_hat @ k_hat^T  (+bias +mask) ----
    const float* biasH = bias16 + (size_t)h * NTOK * NTOK;
    const float* maskW = mask + (size_t)(b_ & (NW - 1)) * NTOK * NTOK;
    for (int t = wave; t < 81; t += 8) {
        const int ti = t / 9, tj = t % 9;
        v8f c = {};
#pragma unroll
        for (int kk = 0; kk < HDIM; kk += 4) {
            v2f a, bfr;
            a.x = qs[(ti * 16 + m16) * SQ + kk + koff];
            a.y = qs[(ti * 16 + m16) * SQ + kk + koff + 1];
            bfr.x = ks[(tj * 16 + m16) * SQ + kk + koff];
            bfr.y = ks[(tj * 16 + m16) * SQ + kk + koff + 1];
            c = __builtin_amdgcn_wmma_f32_16x16x4_f32(false, a, false, bfr,
                                                      (short)0, c, false, false);
        }
#pragma unroll
        for (int r = 0; r < 8; ++r) {
            int row = ti * 16 + half8 + r;
            int colc = tj * 16 + m16;
            S[row * SP + colc] =
                c[r] + biasH[row * NTOK + colc] + maskW[row * NTOK + colc];
        }
    }
    __syncthreads();

    // ---- row softmax ----
    for (int r = wave; r < NTOK; r += 8) {
        float mx = -3.4e38f;
        for (int cc = lane; cc < NTOK; cc += 32) mx = fmaxf(mx, S[r * SP + cc]);
#pragma unroll
        for (int off = 16; off; off >>= 1) mx = fmaxf(mx, __shfl_xor(mx, off, 32));
        float sum = 0.f;
        for (int cc = lane; cc < NTOK; cc += 32) {
            float e = __expf(S[r * SP + cc] - mx);
            S[r * SP + cc] = e;
            sum += e;
        }
#pragma unroll
        for (int off = 16; off; off >>= 1) sum += __shfl_xor(sum, off, 32);
        float inv = 1.0f / sum;
        for (int cc = lane; cc < NTOK; cc += 32) S[r * SP + cc] *= inv;
    }
    __syncthreads();

    // ---- O = P @ V ----
    for (int t = wave; t < 18; t += 8) {
        const int ti = t >> 1, tj = t & 1;
        v8f c = {};
        for (int kk = 0; kk < NTOK; kk += 4) {
            v2f a, bfr;
            a.x = S[(ti * 16 + m16) * SP + kk + koff];
            a.y = S[(ti * 16 + m16) * SP + kk + koff + 1];
            bfr.x = vs[(kk + koff) * HDIM + tj * 16 + m16];
            bfr.y = vs[(kk + koff + 1) * HDIM + tj * 16 + m16];
            c = __builtin_amdgcn_wmma_f32_16x16x4_f32(false, a, false, bfr,
                                                      (short)0, c, false, false);
        }
#pragma unroll
        for (int r = 0; r < 8; ++r) {
            int row = ti * 16 + half8 + r;     // token
            int cd = tj * 16 + m16;            // head-dim col
            attn_out[((size_t)b_ * NTOK + row) * DIM + h * HDIM + cd] = c[r];
        }
    }
}

// ---------------------------------------------------------------------------
// Kernel 5: output projection.  out = attn_out @ w_proj^T + b_proj
// ---------------------------------------------------------------------------
__global__ void proj_gemm_kernel(const float* __restrict__ a_in,
                                 const float* __restrict__ w_proj,
                                 const float* __restrict__ b_proj,
                                 float* __restrict__ out) {
    const int wave = threadIdx.x >> 5;
    const int lane = threadIdx.x & 31;
    const int m16 = lane & 15;
    const int koff = (lane >> 4) << 1;
    const int half8 = (lane >> 4) << 3;

    const int Mbase = blockIdx.y * 16;
    const int Nbase = (blockIdx.x * 8 + wave) * 16;
    const int col = Nbase + m16;

    const float* arow = a_in + (size_t)(Mbase + m16) * DIM;
    const float* brow = w_proj + (size_t)col * DIM;

    v8f c = {};
    for (int kk = 0; kk < DIM; kk += 4) {
        v2f a, b;
        a.x = arow[kk + koff];
        a.y = arow[kk + koff + 1];
        b.x = brow[kk + koff];
        b.y = brow[kk + koff + 1];
        c = __builtin_amdgcn_wmma_f32_16x16x4_f32(false, a, false, b,
                                                  (short)0, c, false, false);
    }
    const float bias = b_proj[col];
#pragma unroll
    for (int r = 0; r < 8; ++r) {
        int row = Mbase + half8 + r;
        out[(size_t)row * DIM + col] = c[r] + bias;
    }
}

// ---------------------------------------------------------------------------
// Launch
// ---------------------------------------------------------------------------
extern "C" void kernel_launch(void* const* d_in, const int* in_sizes, int n_in,
                              void* d_out, int out_size, void* d_ws, size_t ws_size,
                              hipStream_t stream) {
    const float* x = (const float*)d_in[0];
    const float* mask = (const float*)d_in[1];
    const float* w_qkv = (const float*)d_in[2];
    const float* q_bias = (const float*)d_in[3];
    const float* v_bias = (const float*)d_in[4];
    const float* logit_scale = (const float*)d_in[5];
    const float* w1 = (const float*)d_in[6];
    const float* b1 = (const float*)d_in[7];
    const float* w2 = (const float*)d_in[8];
    const float* w_proj = (const float*)d_in[9];
    const float* b_proj = (const float*)d_in[10];
    float* out = (float*)d_out;

    // workspace layout (floats)
    const size_t SZ_TBL = (size_t)TBL * TBL * HEADS;            // 8464
    const size_t SZ_BIAS = (size_t)HEADS * NTOK * NTOK;         // 331776
    const size_t SZ_QKV = (size_t)BWIN * HEADS * NTOK * HDIM;   // 37748736
    float* ws = (float*)d_ws;
    float* table = ws;
    float* bias16 = table + SZ_TBL;
    float* qb = bias16 + SZ_BIAS;
    float* kb = qb + SZ_QKV;
    float* vb = kb + SZ_QKV;
    float* ao = vb + SZ_QKV;

    cpb_table_kernel<<<TBL * TBL, 256, 0, stream>>>(w1, b1, w2, table);

    {
        int tot = HEADS * NTOK * NTOK;
        cpb_bias_kernel<<<(tot + 255) / 256, 256, 0, stream>>>(table, bias16);
    }

    // QKV GEMM: 4608 M-tiles x 96 N-tiles, 8 waves (tiles) per block
    qkv_gemm_kernel<<<dim3(96 / 8, (BWIN * NTOK) / 16), 256, 0, stream>>>(
        x, w_qkv, q_bias, v_bias, qb, kb, vb);

    // fused attention: one block per (b,h)
    {
        size_t shmem = (size_t)(2 * NTOK * SQ + NTOK * HDIM + NTOK * SP) * sizeof(float);
        attention_kernel<<<BWIN * HEADS, 256, shmem, stream>>>(
            qb, kb, vb, bias16, mask, logit_scale, ao);
    }

    // output projection: 4608 M-tiles x 32 N-tiles
    proj_gemm_kernel<<<dim3(4, (BWIN * NTOK) / 16), 256, 0, stream>>>(
        ao, w_proj, b_proj, out);
}